// CRF_63307817943767
// MI455X (gfx1250) — compile-verified
//
#include <hip/hip_runtime.h>
#include <hip/hip_bf16.h>
#include <stdint.h>

typedef float v2f  __attribute__((ext_vector_type(2)));
typedef float v8f  __attribute__((ext_vector_type(8)));
typedef float f4   __attribute__((ext_vector_type(4)));
typedef unsigned int u32x4 __attribute__((ext_vector_type(4)));
typedef int i32x8 __attribute__((ext_vector_type(8)));
typedef int i32x4 __attribute__((ext_vector_type(4)));

#define Bb 64
#define Ss 1024
#define Dd 1024
#define NT 33          // logits / CRF states (BOS column is numerically dead)
#define NN 34          // full transition dim

// ---- GEMM tiling ----
#define KC 32          // K-chunk
#define ROWS_PER_WG 128
#define ASTRIDE 34     // KC + 2 pad dwords (TDM pad) -> conflict-free ds_load_b64
#define WSTRIDE 34     // W natural row stride (floats)
#define L2E 1.4426950408889634f
#define LN2 0.69314718055994531f

// Issue one TDM 2D tile load (Global -> LDS). clang-23 6-arg builtin form.
// g1w0 packs data_size=4B (+ optional LDS pad config). tensor_dims set huge
// (no OOB ever triggers; our tiles are exact).
__device__ __forceinline__ void tdm_load(unsigned lds_off, const float* gptr,
                                         unsigned g1w0, unsigned tile_d0,
                                         unsigned tile_d1, unsigned stride0) {
  unsigned long long ga = (unsigned long long)(const void*)gptr;
  u32x4 g0;
  g0.x = 1u;                                   // count=1, user descriptor
  g0.y = lds_off;                              // lds_addr (bytes)
  g0.z = (unsigned)(ga & 0xFFFFFFFFull);       // global_addr[31:0]
  g0.w = (unsigned)(ga >> 32) | 0x80000000u;   // global_addr[56:32] | type=2
  i32x8 g1;
  g1[0] = (int)g1w0;                           // mask=0,data_size=2(4B),pad cfg
  g1[1] = 0;                                   // tensor_dim0 = 0x40000000 (lo16=0)
  g1[2] = 0x4000;                              // td0 hi16 | td1 lo16(=0)
  g1[3] = (int)(0x4000u | (tile_d0 << 16));    // td1 hi16 | tile_dim0
  g1[4] = (int)tile_d1;                        // tile_dim1 | tile_dim2=0
  g1[5] = (int)stride0;                        // tensor_dim0_stride lo32
  g1[6] = 0;
  g1[7] = 0;
  i32x4 z4 = {0, 0, 0, 0};
  i32x8 z8 = {0, 0, 0, 0, 0, 0, 0, 0};
  __builtin_amdgcn_tensor_load_to_lds(g0, g1, z4, z4, z8, 0);
}

// g1 word0 flags
#define G1_W     0x00020000u                                   // 4B elems, no pad
#define G1_A     (0x00020000u | (1u<<20) | (4u<<22) | (1u<<25)) // pad: every 32 dw, +2 dw

__global__ __launch_bounds__(256)
void crf_gemm_kernel(const float* __restrict__ em, const float* __restrict__ W,
                     const float* __restrict__ bias, float* __restrict__ out) {
  __shared__ float As[2][ROWS_PER_WG * ASTRIDE];   // 2 x 128 x (32+2pad)
  __shared__ float Ws[2][1152];                    // 2 x (32x34 + slack)
  const int tid  = threadIdx.x;
  const int wave = tid >> 5;
  const int lane = tid & 31;
  const int nl   = lane & 15;
  const int kb   = (lane >> 4) << 1;
  const int rowBase = blockIdx.x * ROWS_PER_WG;

  v8f acc0 = {}; v8f acc1 = {}; v8f acc2 = {};

  const int nChunks = Dd / KC;   // 32
  if (tid < 32) {   // wave 0 drives the TDM
    tdm_load((unsigned)(unsigned long long)(void*)&As[0][0],
             em + (size_t)rowBase * Dd, G1_A, KC, ROWS_PER_WG, Dd);
    tdm_load((unsigned)(unsigned long long)(void*)&Ws[0][0],
             W, G1_W, KC * NN, 1, KC * NN);
  }
  for (int c = 0; c < nChunks; ++c) {
    const int buf = c & 1;
    if (tid < 32) { __builtin_amdgcn_s_wait_tensorcnt(0); }
    __syncthreads();                       // tile `c` resident for everyone
    if (tid < 32 && (c + 1) < nChunks) {   // prefetch next chunk into other buf
      tdm_load((unsigned)(unsigned long long)(void*)&As[buf ^ 1][0],
               em + (size_t)rowBase * Dd + (size_t)(c + 1) * KC,
               G1_A, KC, ROWS_PER_WG, Dd);
      tdm_load((unsigned)(unsigned long long)(void*)&Ws[buf ^ 1][0],
               W + (size_t)(c + 1) * KC * NN, G1_W, KC * NN, 1, KC * NN);
    }
    const float* A  = &As[buf][(wave * 16 + nl) * ASTRIDE + kb];
    const float* Wl = &Ws[buf][0];
#pragma unroll
    for (int k = 0; k < KC; k += 4) {
      v2f a = *(const v2f*)(A + k);
      const int r0 = (k + kb) * WSTRIDE, r1 = (k + kb + 1) * WSTRIDE;
      v2f b0, b1, b2;
      b0.x = Wl[r0 + nl];      b0.y = Wl[r1 + nl];
      b1.x = Wl[r0 + 16 + nl]; b1.y = Wl[r1 + 16 + nl];
      b2.x = Wl[r0 + 32 + nl]; b2.y = Wl[r1 + 32 + nl];
      acc0 = __builtin_amdgcn_wmma_f32_16x16x4_f32(false, a, false, b0, (short)0, acc0, false, false);
      acc1 = __builtin_amdgcn_wmma_f32_16x16x4_f32(false, a, false, b1, (short)0, acc1, false, false);
      acc2 = __builtin_amdgcn_wmma_f32_16x16x4_f32(false, a, false, b2, (short)0, acc2, false, false);
    }
    __syncthreads();                       // done reading buf before overwrite
  }
  // epilogue: +bias, store logits cols 0..32 only
  const int rsel = (lane >> 4) * 8;
  const float bv0 = bias[nl];
  const float bv1 = bias[16 + nl];
  const float bv2 = (nl == 0) ? bias[32] : 0.f;
#pragma unroll
  for (int r = 0; r < 8; ++r) {
    const size_t m = (size_t)rowBase + wave * 16 + r + rsel;
    float* op = out + 1 + m * NT;
    op[nl] = acc0[r] + bv0;
    op[16 + nl] = acc1[r] + bv1;
    if (nl == 0) op[32] = acc2[r] + bv2;
  }
}

__global__ __launch_bounds__(64)
void crf_scan_kernel(const float* __restrict__ logits, const int* __restrict__ tags,
                     const float* __restrict__ mask, const float* __restrict__ T,
                     float* __restrict__ partials) {
  const int b = blockIdx.x;
  const int tid = threadIdx.x;
  __shared__ float Tl[NN * 35];
  __shared__ __align__(16) float al[2][36];
  __shared__ float red[64];
  for (int idx = tid; idx < NN * NN; idx += 64)
    Tl[(idx / NN) * 35 + (idx % NN)] = T[idx];
  if (tid < 3) { al[0][NT + tid] = -1e30f; al[1][NT + tid] = -1e30f; }
  __syncthreads();

  const int j = tid;
  const bool act = (j < NT);
  float Ecol[NT];
#pragma unroll
  for (int i = 0; i < NT; ++i)
    Ecol[i] = act ? __builtin_amdgcn_exp2f(Tl[i * 35 + j] * L2E) : 0.f;

  const float* lg = logits + (size_t)b * Ss * NT;
  // gold-path score (parallel over t, strided)
  float sc = 0.f;
  for (int t = 1 + tid; t < Ss; t += 64) {
    const int tg = tags[b * Ss + t];
    const int tp = tags[b * Ss + t - 1];
    sc += mask[b * Ss + t] * (lg[t * NT + tg] + Tl[tp * 35 + tg]);
  }
  if (tid == 0) {
    const int f = tags[b * Ss];
    sc += Tl[NT * 35 + f] + lg[f];     // transitions[BOS, first] + em[0, first]
  }
  red[tid] = sc;

  // forward scan, 33 live states, 1 barrier/step, double-buffered alphas
  float a = -1e30f;
  if (act) { a = Tl[NT * 35 + j] + lg[j]; al[0][j] = a; }
  float em_next = act ? lg[1 * NT + j] : 0.f;
  for (int t = 1; t < Ss; ++t) {
    __syncthreads();
    const float* ar = al[(t - 1) & 1];
    f4 q[9];
#pragma unroll
    for (int r = 0; r < 9; ++r) q[r] = ((const f4*)ar)[r];
    float m = -1e30f;
#pragma unroll
    for (int i = 0; i < NT; ++i) m = fmaxf(m, q[i >> 2][i & 3]);
    float s = 0.f;
#pragma unroll
    for (int i = 0; i < NT; ++i)
      s = fmaf(__builtin_amdgcn_exp2f((q[i >> 2][i & 3] - m) * L2E), Ecol[i], s);
    const float anew = m + __builtin_amdgcn_logf(s) * LN2 + em_next;
    const float mt = mask[b * Ss + t];
    a = act ? (mt * anew + (1.f - mt) * a) : a;
    if (t + 1 < Ss) em_next = act ? lg[(t + 1) * NT + j] : 0.f;
    if (act) al[t & 1][j] = a;
  }
  __syncthreads();
  if (tid == 0) {
    const float* af = al[(Ss - 1) & 1];
    float m = -1e30f;
    for (int i = 0; i < NT; ++i) m = fmaxf(m, af[i]);
    float s = 0.f;
    for (int i = 0; i < NT; ++i) s += __builtin_amdgcn_exp2f((af[i] - m) * L2E);
    const float part = m + __builtin_amdgcn_logf(s) * LN2;
    float scoreSum = 0.f;
    for (int k = 0; k < 64; ++k) scoreSum += red[k];
    partials[b] = part - scoreSum;       // per-batch (partition - score)
  }
}

__global__ void crf_loss_kernel(const float* __restrict__ partials, float* __restrict__ out) {
  if (threadIdx.x == 0 && blockIdx.x == 0) {
    float s = 0.f;
    for (int i = 0; i < Bb; ++i) s += partials[i];   // fixed order -> deterministic
    out[0] = s;
  }
}

extern "C" void kernel_launch(void* const* d_in, const int* in_sizes, int n_in,
                              void* d_out, int out_size, void* d_ws, size_t ws_size,
                              hipStream_t stream) {
  const float* em   = (const float*)d_in[0];
  const int*   tags = (const int*)d_in[1];
  const float* mask = (const float*)d_in[2];
  const float* W    = (const float*)d_in[3];
  const float* bias = (const float*)d_in[4];
  const float* T    = (const float*)d_in[5];
  float* out = (float*)d_out;
  float* partials = (float*)d_ws;
  (void)in_sizes; (void)n_in; (void)out_size; (void)ws_size;

  crf_gemm_kernel<<<(Bb * Ss) / ROWS_PER_WG, 256, 0, stream>>>(em, W, bias, out);
  crf_scan_kernel<<<Bb, 64, 0, stream>>>(out + 1, tags, mask, T, partials);
  crf_loss_kernel<<<1, 1, 0, stream>>>(partials, out);
}